// LSTM_6201932776128
// MI455X (gfx1250) — compile-verified
//
#include <hip/hip_runtime.h>
#include <math.h>

// LSTM reference reduction: only the T-1 slice of the carried (T,B,H) state
// reaches the output, so we run a single (B=64,H=768) LSTM cell whose input
// projection xg = X[:,T-1,:] @ W_g + b_g is fixed, iterated T=64 times.
#define TT 64
#define BB 64
#define DD 768
#define HH 768

typedef __attribute__((ext_vector_type(2))) float v2f;
typedef __attribute__((ext_vector_type(4))) float v4f;
typedef __attribute__((ext_vector_type(8))) float v8f;

__device__ __forceinline__ v8f wmma_f32_k4(v2f a, v2f b, v8f c) {
  // D = A(16x4 f32) * B(4x16 f32) + C(16x16 f32)
  return __builtin_amdgcn_wmma_f32_16x16x4_f32(
      /*neg_a=*/false, a, /*neg_b=*/false, b,
      /*c_mod=*/(short)0, c, /*reuse_a=*/false, /*reuse_b=*/false);
}

__device__ __forceinline__ float sigmoidf_(float x) {
  return 1.0f / (1.0f + expf(-x));
}

// ---------------------------------------------------------------------------
// Repack the four recurrent matrices U_g (H x H, row major) into WMMA
// B-fragment lane order so the step kernel can fetch each lane's fragment for
// TWO k-steps (K=8) as one contiguous b128:
//   upk[(((g*48 + nt)*96 + kp)*32 + lane)*4 + {0..3}] =
//     U_g[(8*kp + kb + 0)*H + n], U_g[(8*kp + kb + 1)*H + n],
//     U_g[(8*kp + kb + 4)*H + n], U_g[(8*kp + kb + 5)*H + n]
//   with n = nt*16 + (lane&15), kb = (lane>>4)*2.
// ---------------------------------------------------------------------------
__global__ void repack_u_kernel(const float* __restrict__ u0, const float* __restrict__ u1,
                                const float* __restrict__ u2, const float* __restrict__ u3,
                                float* __restrict__ upk) {
  const int tid  = blockIdx.x * blockDim.x + threadIdx.x; // 0 .. 4*48*96*32-1
  const int lane = tid & 31;
  const int kp   = (tid >> 5) % 96;
  const int nt   = ((tid >> 5) / 96) % 48;
  const int g    = tid / (32 * 96 * 48);

  const float* U = (g == 0) ? u0 : (g == 1) ? u1 : (g == 2) ? u2 : u3;
  const int n  = nt * 16 + (lane & 15);
  const int kb = (lane >> 4) * 2;
  const int k0 = kp * 8;

  v4f v;
  v.x = U[(size_t)(k0 + kb + 0) * HH + n];
  v.y = U[(size_t)(k0 + kb + 1) * HH + n];
  v.z = U[(size_t)(k0 + kb + 4) * HH + n];
  v.w = U[(size_t)(k0 + kb + 5) * HH + n];
  *(v4f*)(upk + (size_t)tid * 4) = v;     // coalesced b128 store
}

// ---------------------------------------------------------------------------
// xg[g][b][n] = sum_d X[b, T-1, d] * W_g[d, n] + bias_g[n]
// One wave per 16x16 tile per gate: 4 gates * 4 mtiles * 48 ntiles = 768 waves.
// Runs once; left in the simple (unpacked-W) form.
// ---------------------------------------------------------------------------
__global__ void lstm_proj_kernel(const float* __restrict__ X,
                                 const float* __restrict__ w0, const float* __restrict__ w1,
                                 const float* __restrict__ w2, const float* __restrict__ w3,
                                 const float* __restrict__ b0, const float* __restrict__ b1,
                                 const float* __restrict__ b2, const float* __restrict__ b3,
                                 float* __restrict__ xg) {
  const int wave = (blockIdx.x * blockDim.x + threadIdx.x) >> 5;
  const int lane = threadIdx.x & 31;
  const int gate = wave / 192;
  const int rem  = wave % 192;
  const int mt = rem / 48;
  const int nt = rem % 48;

  const float* W    = (gate == 0) ? w0 : (gate == 1) ? w1 : (gate == 2) ? w2 : w3;
  const float* bias = (gate == 0) ? b0 : (gate == 1) ? b1 : (gate == 2) ? b2 : b3;

  const int m  = mt * 16 + (lane & 15);          // A fragment row
  const int n  = nt * 16 + (lane & 15);          // B fragment col
  const int kb = (lane >> 4) * 2;                // K pair held by this lane
  const float* arow = X + (size_t)m * TT * DD + (size_t)(TT - 1) * DD;

  v8f acc = {};
#pragma unroll 4
  for (int k = 0; k < DD; k += 4) {
    v2f a = *(const v2f*)(arow + k + kb);
    v2f b;
    b.x = W[(size_t)(k + kb) * HH + n];
    b.y = W[(size_t)(k + kb + 1) * HH + n];
    acc = wmma_f32_k4(a, b, acc);
  }

  const int mhalf = (lane >> 4) * 8;             // C/D layout: vgpr r -> M = r + half*8
  const float bn = bias[n];
#pragma unroll
  for (int r = 0; r < 8; ++r) {
    const int mr = mt * 16 + mhalf + r;
    xg[(size_t)gate * BB * HH + (size_t)mr * HH + n] = acc[r] + bn;
  }
}

// ---------------------------------------------------------------------------
// One recurrence step. Each wave owns a 32x16 output strip (two M-tiles) for
// all four gates: 8 f32 accumulators (64 VGPRs). Per K=8 iteration:
//   4x global_load_b64 (A fragments) + 4x global_load_b128 (packed B) and
//   16x v_wmma_f32_16x16x4_f32. Epilogue fuses gate activations, mask blend
//   and the c/h update. 2 mgroups * 48 ntiles = 96 waves.
// ---------------------------------------------------------------------------
__global__ void lstm_step_kernel(const float* __restrict__ hprev, const float* __restrict__ cprev,
                                 float* __restrict__ hnext, float* __restrict__ cnext,
                                 const float* __restrict__ upk,
                                 const float* __restrict__ xg, const float* __restrict__ mask) {
  const int wave = (blockIdx.x * blockDim.x + threadIdx.x) >> 5;
  const int lane = threadIdx.x & 31;
  const int mt2 = wave / 48;                     // 0..1 -> M rows [mt2*32, mt2*32+32)
  const int nt  = wave % 48;

  const int lrow = lane & 15;
  const int kb   = (lane >> 4) * 2;
  const int m0   = mt2 * 32 + lrow;
  const int m1   = m0 + 16;
  const int n    = nt * 16 + lrow;

  const float* a0p = hprev + (size_t)m0 * HH + kb;
  const float* a1p = hprev + (size_t)m1 * HH + kb;
  // packed-B base for (gate g, nt), this lane: stride between gates = 48*96*128
  const float* bp0 = upk + ((size_t)(0 * 48 + nt) * 96 * 32 + lane) * 4;
  const float* bp1 = upk + ((size_t)(1 * 48 + nt) * 96 * 32 + lane) * 4;
  const float* bp2 = upk + ((size_t)(2 * 48 + nt) * 96 * 32 + lane) * 4;
  const float* bp3 = upk + ((size_t)(3 * 48 + nt) * 96 * 32 + lane) * 4;

  v8f acc[4][2] = {};
#pragma unroll 2
  for (int kp = 0; kp < 96; ++kp) {
    const int ka = kp * 8;
    v2f a0a = *(const v2f*)(a0p + ka);
    v2f a0b = *(const v2f*)(a0p + ka + 4);
    v2f a1a = *(const v2f*)(a1p + ka);
    v2f a1b = *(const v2f*)(a1p + ka + 4);

    v4f b0 = *(const v4f*)(bp0 + (size_t)kp * 128);
    v4f b1 = *(const v4f*)(bp1 + (size_t)kp * 128);
    v4f b2 = *(const v4f*)(bp2 + (size_t)kp * 128);
    v4f b3 = *(const v4f*)(bp3 + (size_t)kp * 128);

    v2f b0lo = __builtin_shufflevector(b0, b0, 0, 1), b0hi = __builtin_shufflevector(b0, b0, 2, 3);
    v2f b1lo = __builtin_shufflevector(b1, b1, 0, 1), b1hi = __builtin_shufflevector(b1, b1, 2, 3);
    v2f b2lo = __builtin_shufflevector(b2, b2, 0, 1), b2hi = __builtin_shufflevector(b2, b2, 2, 3);
    v2f b3lo = __builtin_shufflevector(b3, b3, 0, 1), b3hi = __builtin_shufflevector(b3, b3, 2, 3);

    acc[0][0] = wmma_f32_k4(a0a, b0lo, acc[0][0]);
    acc[0][0] = wmma_f32_k4(a0b, b0hi, acc[0][0]);
    acc[0][1] = wmma_f32_k4(a1a, b0lo, acc[0][1]);
    acc[0][1] = wmma_f32_k4(a1b, b0hi, acc[0][1]);

    acc[1][0] = wmma_f32_k4(a0a, b1lo, acc[1][0]);
    acc[1][0] = wmma_f32_k4(a0b, b1hi, acc[1][0]);
    acc[1][1] = wmma_f32_k4(a1a, b1lo, acc[1][1]);
    acc[1][1] = wmma_f32_k4(a1b, b1hi, acc[1][1]);

    acc[2][0] = wmma_f32_k4(a0a, b2lo, acc[2][0]);
    acc[2][0] = wmma_f32_k4(a0b, b2hi, acc[2][0]);
    acc[2][1] = wmma_f32_k4(a1a, b2lo, acc[2][1]);
    acc[2][1] = wmma_f32_k4(a1b, b2hi, acc[2][1]);

    acc[3][0] = wmma_f32_k4(a0a, b3lo, acc[3][0]);
    acc[3][0] = wmma_f32_k4(a0b, b3hi, acc[3][0]);
    acc[3][1] = wmma_f32_k4(a1a, b3lo, acc[3][1]);
    acc[3][1] = wmma_f32_k4(a1b, b3hi, acc[3][1]);
  }

  const int mhalf = (lane >> 4) * 8;
#pragma unroll
  for (int t = 0; t < 2; ++t) {
    const int mbase = mt2 * 32 + t * 16 + mhalf;
#pragma unroll
    for (int r = 0; r < 8; ++r) {
      const int mr = mbase + r;
      const size_t idx = (size_t)mr * HH + n;
      const float pi = acc[0][t][r] + xg[idx];
      const float pf = acc[1][t][r] + xg[(size_t)1 * BB * HH + idx];
      const float pc = acc[2][t][r] + xg[(size_t)2 * BB * HH + idx];
      const float po = acc[3][t][r] + xg[(size_t)3 * BB * HH + idx];

      const float i_t = tanhf(pi);               // reference uses tanh for input gate
      const float f_t = sigmoidf_(pf);
      const float c_t = f_t * cprev[idx] + i_t * tanhf(pc);
      const float o_t = sigmoidf_(po);
      const float h_t = o_t * tanhf(c_t);

      const float mm = mask[(size_t)mr * TT + (TT - 1)];
      hnext[idx] = (1.0f - mm) * hprev[idx] + mm * h_t;
      cnext[idx] = (1.0f - mm) * cprev[idx] + mm * c_t;
    }
  }
}

extern "C" void kernel_launch(void* const* d_in, const int* in_sizes, int n_in,
                              void* d_out, int out_size, void* d_ws, size_t ws_size,
                              hipStream_t stream) {
  (void)in_sizes; (void)n_in; (void)out_size; (void)ws_size;
  const float* X    = (const float*)d_in[0];
  const float* mask = (const float*)d_in[1];
  const float* w0 = (const float*)d_in[2];
  const float* w1 = (const float*)d_in[3];
  const float* w2 = (const float*)d_in[4];
  const float* w3 = (const float*)d_in[5];
  const float* u0 = (const float*)d_in[6];
  const float* u1 = (const float*)d_in[7];
  const float* u2 = (const float*)d_in[8];
  const float* u3 = (const float*)d_in[9];
  const float* b0 = (const float*)d_in[10];
  const float* b1 = (const float*)d_in[11];
  const float* b2 = (const float*)d_in[12];
  const float* b3 = (const float*)d_in[13];
  float* out = (float*)d_out;

  // Workspace (floats): xg[4*B*H] | hA | cA | hB | cB | upk[4*H*H]  (~11 MB)
  float* xg  = (float*)d_ws;
  float* hA  = xg + (size_t)4 * BB * HH;
  float* cA  = hA + (size_t)BB * HH;
  float* hB  = cA + (size_t)BB * HH;
  float* cB  = hB + (size_t)BB * HH;
  float* upk = cB + (size_t)BB * HH;

  // h0 = c0 = 0 (hA and cA are adjacent -> single async memset; capture-safe)
  hipMemsetAsync(hA, 0, (size_t)2 * BB * HH * sizeof(float), stream);

  // One-time repack of U into WMMA B-fragment lane order (L2-resident after).
  repack_u_kernel<<<(4 * 48 * 96 * 32) / 256, 256, 0, stream>>>(u0, u1, u2, u3, upk);

  // Input projection for the T-1 slice: 768 waves (96 blocks x 8 waves).
  lstm_proj_kernel<<<96, 256, 0, stream>>>(X, w0, w1, w2, w3, b0, b1, b2, b3, xg);

  // 64 recurrence steps, double-buffered; last step writes h directly to d_out.
  for (int s = 0; s < TT; ++s) {
    const float* hp = (s & 1) ? hB : hA;
    const float* cp = (s & 1) ? cB : cA;
    float* hn = (s & 1) ? hA : hB;
    float* cn = (s & 1) ? cA : cB;
    if (s == TT - 1) hn = out;
    lstm_step_kernel<<<12, 256, 0, stream>>>(hp, cp, hn, cn, upk, xg, mask);
  }
}